// ExpertLoRACollection_6305011990868
// MI455X (gfx1250) — compile-verified
//
#include <hip/hip_runtime.h>

typedef __attribute__((ext_vector_type(2))) float v2f;
typedef __attribute__((ext_vector_type(8))) float v8f;

// Problem constants (fixed by setup_inputs)
constexpr int Bn = 8;      // batch
constexpr int S  = 2048;   // sequence
constexpr int H  = 3584;   // hidden
constexpr int R  = 16;     // lora rank
constexpr int Kx = 2;      // top-k experts
constexpr float SCALING = 32.0f / 16.0f;

// ---------------------------------------------------------------------------
// Pass 1: low[b,k,s,r] = w[b,k]*SCALING * sum_h x[b,s,h] * loraA[e(b,k),r,h]
// One wave handles one 16-row s-tile for one b, BOTH experts (x read once).
// WMMA f32 16x16x4: A = x (M=s, K=h), B = loraA (K=h, N=r).
// ---------------------------------------------------------------------------
__global__ __launch_bounds__(256) void lora_down_kernel(
    const float* __restrict__ x,      // [B,S,H]
    const float* __restrict__ loraA,  // [E,R,H]
    const float* __restrict__ w,      // [B,K]
    const int*   __restrict__ idx,    // [B,K]
    float*       __restrict__ low)    // [B,K,S,R]
{
  const int lane = threadIdx.x & 31;
  const int wave = threadIdx.x >> 5;
  const int tile = blockIdx.x * 8 + wave;         // Bn * (S/16) = 1024 tiles
  const int b    = tile >> 7;                     // / (S/16 = 128)
  const int s0   = (tile & 127) << 4;

  const int e0 = idx[b * Kx + 0];
  const int e1 = idx[b * Kx + 1];
  const float w0 = w[b * Kx + 0] * SCALING;
  const float w1 = w[b * Kx + 1] * SCALING;

  const int lmod = lane & 15;
  const int lhi  = lane >> 4;
  const int hoff = lhi << 1;                      // lanes 16-31 carry K=2,3

  const float* xa  = x     + ((size_t)b  * S + s0 + lmod) * H + hoff;
  const float* a0p = loraA + ((size_t)e0 * R + lmod) * H + hoff;
  const float* a1p = loraA + ((size_t)e1 * R + lmod) * H + hoff;

  // Two accumulator chains per expert for ILP on the WMMA accumulate chain.
  v8f c0a = {}, c0b = {}, c1a = {}, c1b = {};

  #pragma unroll 4
  for (int h = 0; h < H; h += 8) {
    v2f av0 = *(const v2f*)(xa  + h);
    v2f av1 = *(const v2f*)(xa  + h + 4);
    v2f b00 = *(const v2f*)(a0p + h);
    v2f b01 = *(const v2f*)(a0p + h + 4);
    v2f b10 = *(const v2f*)(a1p + h);
    v2f b11 = *(const v2f*)(a1p + h + 4);
    c0a = __builtin_amdgcn_wmma_f32_16x16x4_f32(false, av0, false, b00,
                                                (short)0, c0a, false, false);
    c1a = __builtin_amdgcn_wmma_f32_16x16x4_f32(false, av0, false, b10,
                                                (short)0, c1a, false, false);
    c0b = __builtin_amdgcn_wmma_f32_16x16x4_f32(false, av1, false, b01,
                                                (short)0, c0b, false, false);
    c1b = __builtin_amdgcn_wmma_f32_16x16x4_f32(false, av1, false, b11,
                                                (short)0, c1b, false, false);
  }

  // D layout: VGPR j -> row s0 + j + 8*lhi, col r = lmod
  float* o0 = low + (((size_t)b * Kx + 0) * S + s0 + lhi * 8) * R + lmod;
  float* o1 = low + (((size_t)b * Kx + 1) * S + s0 + lhi * 8) * R + lmod;
  #pragma unroll
  for (int j = 0; j < 8; ++j) {
    o0[j * R] = (c0a[j] + c0b[j]) * w0;
    o1[j * R] = (c1a[j] + c1b[j]) * w1;
  }
}

// ---------------------------------------------------------------------------
// Pass 2: out[b,s,h] = base[b,s,h] + sum_{k,r} low[b,k,s,r] * loraB[e(b,k),h,r]
// Per wave: one [16s x 16h] tile. WMMA K-dim = concatenated (k,r) of size 32,
// 8 steps of 16x16x4, C seeded with the base tile so D is the final output.
// ---------------------------------------------------------------------------
__global__ __launch_bounds__(256) void lora_up_kernel(
    const float* __restrict__ low,    // [B,K,S,R]
    const float* __restrict__ loraB,  // [E,H,R]
    const float* __restrict__ base,   // [B,S,H]
    const int*   __restrict__ idx,    // [B,K]
    float*       __restrict__ out)    // [B,S,H]
{
  constexpr int HT = H / 16;                       // 224
  const int lane = threadIdx.x & 31;
  const int wave = threadIdx.x >> 5;
  const int tile = blockIdx.x * 8 + wave;          // Bn*(S/16)*(H/16)
  const int b    = tile / (128 * HT);
  const int rem  = tile % (128 * HT);
  const int s0   = (rem / HT) << 4;
  const int h0   = (rem % HT) << 4;

  const int e0 = idx[b * Kx + 0];
  const int e1 = idx[b * Kx + 1];

  const int lmod = lane & 15;
  const int lhi  = lane >> 4;

  // C/D layout: VGPR j -> row s0 + j + 8*lhi, col h0 + lmod
  const float* bp = base + ((size_t)b * S + s0 + lhi * 8) * H + h0 + lmod;
  float*       op = out  + ((size_t)b * S + s0 + lhi * 8) * H + h0 + lmod;

  v8f c;
  #pragma unroll
  for (int j = 0; j < 8; ++j) c[j] = bp[(size_t)j * H];

  const float* lowb = low + (size_t)b * Kx * S * R;

  #pragma unroll
  for (int kk = 0; kk < Kx * R; kk += 4) {
    const int k    = kk >> 4;                      // which expert slot
    const int e    = (k == 0) ? e0 : e1;
    const int roff = (kk & 15) + (lhi << 1);       // lanes 16-31: K=+2,+3
    // A: low rows (M=s), K = (k,r) concat
    v2f av = *(const v2f*)(lowb + ((size_t)k * S + s0 + lmod) * R + roff);
    // B: loraB (K = r within expert, N = h)
    v2f bv = *(const v2f*)(loraB + ((size_t)e * H + h0 + lmod) * R + roff);
    c = __builtin_amdgcn_wmma_f32_16x16x4_f32(false, av, false, bv,
                                              (short)0, c, false, false);
  }

  #pragma unroll
  for (int j = 0; j < 8; ++j) op[(size_t)j * H] = c[j];
}

extern "C" void kernel_launch(void* const* d_in, const int* in_sizes, int n_in,
                              void* d_out, int out_size, void* d_ws, size_t ws_size,
                              hipStream_t stream) {
  const float* x      = (const float*)d_in[0];   // [B,S,H]
  const float* base   = (const float*)d_in[1];   // [B,S,H]
  const float* loraA  = (const float*)d_in[2];   // [E,R,H]
  const float* loraB  = (const float*)d_in[3];   // [E,H,R]
  const float* topw   = (const float*)d_in[4];   // [B,K]
  const int*   topidx = (const int*)d_in[5];     // [B,K]
  float*       out    = (float*)d_out;
  float*       low    = (float*)d_ws;            // [B,K,S,R] = 2 MB

  // Pass 1: 1024 s-tiles, 8 waves (256 threads) per block -> 128 blocks
  lora_down_kernel<<<Bn * (S / 16) / 8, 256, 0, stream>>>(x, loraA, topw,
                                                          topidx, low);
  // Pass 2: 8*128*224 = 229376 tiles / 8 waves -> 28672 blocks
  lora_up_kernel<<<Bn * (S / 16) * (H / 16) / 8, 256, 0, stream>>>(
      low, loraB, base, topidx, out);
}